// get_model_62972810494422
// MI455X (gfx1250) — compile-verified
//
#include <hip/hip_runtime.h>

typedef float v2f __attribute__((ext_vector_type(2)));
typedef float v8f __attribute__((ext_vector_type(8)));

#define S2D 384
#define GHALF 128

// ---------------- init: zero output, init per-batch mins ----------------
__global__ __launch_bounds__(256) void init_kernel(float* __restrict__ out, int out_n,
                                                   int* __restrict__ minbuf, int nmin) {
  int i = blockIdx.x * blockDim.x + threadIdx.x;
  if (i < out_n) out[i] = 0.0f;
  if (i < nmin) minbuf[i] = 0x7FFFFFFF;
}

// ---------------- pass1: elevate (WMMA f32), rank/shift, keys + batch min ----
__global__ __launch_bounds__(256) void pass1_kernel(const float* __restrict__ pc1,
                                                    const float* __restrict__ trans,
                                                    int2* __restrict__ keys,
                                                    int* __restrict__ minbuf,
                                                    int N, int blocksPerBatch) {
  const int b     = blockIdx.x / blocksPerBatch;
  const int npart = blockIdx.x % blocksPerBatch;
  const int tid   = threadIdx.x;
  const int wave  = tid >> 5;
  const int lane  = tid & 31;
  const int m     = lane & 15;
  const bool hiL  = lane >= 16;
  const int n0    = npart * 256 + wave * 32;     // 32 points per wave (2 WMMA tiles of 16)

  const float* T = trans + (size_t)b * 9;
  const float* P = pc1 + (size_t)b * 3 * N;

  // A (16x4 f32): v0 holds K=0 (lanes 0-15) / K=2 (lanes 16-31); v1 holds K=1 / K=3
  v2f a;
  const int ka0 = hiL ? 2 : 0;
  a.x = (m < 3) ? T[m * 3 + ka0] : 0.0f;          // K=0 or K=2
  a.y = (!hiL && m < 3) ? T[m * 3 + 1] : 0.0f;    // K=1 ; K=3 is zero-pad

  // B (4x16 f32), N-index = lane&15: v0 holds K=0 / K=2 ; v1 holds K=1 / K=3
  v2f b0, b1;
  const int kb0 = hiL ? 2 : 0;
  b0.x = P[(size_t)kb0 * N + n0 + m];
  b0.y = hiL ? 0.0f : P[(size_t)1 * N + n0 + m];
  b1.x = P[(size_t)kb0 * N + n0 + 16 + m];
  b1.y = hiL ? 0.0f : P[(size_t)1 * N + n0 + 16 + m];

  v8f cz = {};
  // D[mm][nn] = sum_k A[mm][k]*B[k][nn] ; rows 0..2 = elevated components, col nn = point
  v8f d0 = __builtin_amdgcn_wmma_f32_16x16x4_f32(false, a, false, b0, (short)0, cz, false, false);
  v8f d1 = __builtin_amdgcn_wmma_f32_16x16x4_f32(false, a, false, b1, (short)0, cz, false, false);

  // C/D layout: VGPR v, lanes0-15 -> (M=v, N=lane). Lane L holds elevated[c] of point n0+L
  // (tile0 for L<16; tile1 results live in lanes 0-15 -> shuffle up to lanes 16-31).
  float e[3];
#pragma unroll
  for (int c = 0; c < 3; ++c) {
    float lov = d0[c];
    float hiv = __shfl(d1[c], m, 32);
    e[c] = hiL ? hiv : lov;
  }

  // ---- rank / remainder / shift (faithful to reference, incl. argsort stability) ----
  float g[3], f[3];
#pragma unroll
  for (int c = 0; c < 3; ++c) {
    g[c] = rintf(e[c] / 3.0f) * 3.0f;   // rintf = round-half-even, matches jnp.round
    f[c] = e[c] - g[c];
  }
  int r[3];
  r[0] = (f[1] > f[0]) + (f[2] > f[0]);
  r[1] = (f[0] > f[1]) + (f[2] > f[1]) + (f[0] == f[1]);
  r[2] = (f[0] > f[2]) + (f[1] > f[2]) + (f[0] == f[2]) + (f[1] == f[2]);

  const float rsf = (g[0] + g[1] + g[2]) / 3.0f;  // exact integer in f32
  const int rsi = (int)rsf;                        // trunc, matches astype(int32)
  int gi[3];
#pragma unroll
  for (int c = 0; c < 3; ++c) {
    float rf = (float)r[c];
    bool cond = ((rf >= 3.0f - rsf) && (rsf > 0.0f)) || ((rf < -rsf) && (rsf < 0.0f));
    float sgn = (rsf > 0.0f ? -1.0f : 0.0f) + (rsf < 0.0f ? 1.0f : 0.0f);
    float shf = 3.0f * sgn * (cond ? 1.0f : 0.0f);
    g[c] += shf;
    r[c] += (int)shf + rsi;
    gi[c] = (int)g[c];
  }

  // keys for vertex j=0 are just greedy (CANONICAL[*][0] == 0)
  keys[(size_t)b * N + n0 + lane] = make_int2(gi[0], gi[1]);

  // per-point min over j of key_i = gi + min(CANONICAL row), with JAX gather
  // index semantics: negative -> +dim once, then clamp to [0,2]
  const int rowmin[3] = {0, -1, -2};
  int rn0 = r[0] < 0 ? r[0] + 3 : r[0];
  int rn1 = r[1] < 0 ? r[1] + 3 : r[1];
  rn0 = min(max(rn0, 0), 2);
  rn1 = min(max(rn1, 0), 2);
  const int cand0 = gi[0] + rowmin[rn0];
  const int cand1 = gi[1] + rowmin[rn1];

  __shared__ int sm0, sm1;
  if (tid == 0) { sm0 = 0x7FFFFFFF; sm1 = 0x7FFFFFFF; }
  __syncthreads();
  atomicMin(&sm0, cand0);
  atomicMin(&sm1, cand1);
  __syncthreads();
  if (tid == 0) {
    atomicMin(&minbuf[2 * b + 0], sm0);
    atomicMin(&minbuf[2 * b + 1], sm1);
  }
}

// ---------------- pass2: scatter-add features into the filtered 128x128 grid ----
__global__ __launch_bounds__(256) void pass2_kernel(const float* __restrict__ feat,
                                                    const int2* __restrict__ keys,
                                                    const int* __restrict__ minbuf,
                                                    float* __restrict__ out,
                                                    int N, int B) {
  long long t = (long long)blockIdx.x * blockDim.x + threadIdx.x;
  if (t >= (long long)B * N) return;
  const int b = (int)(t / N);
  const int n = (int)(t % N);
  const int2 k = keys[t];
  const int m0 = minbuf[2 * b + 0];
  const int m1 = minbuf[2 * b + 1];
  const int c0 = k.x - m0;
  const int c1 = k.y - m1;
  if ((unsigned)c0 < (unsigned)S2D && (unsigned)c1 < (unsigned)S2D) {
    const int pp0 = ((-m0) % 3 + 3) % 3;
    const int pp1 = ((-m1) % 3 + 3) % 3;
    int rr = c0 - pp0, cc = c1 - pp1;
    if (rr >= 0 && cc >= 0) {          // key%3==0 guarantees divisibility
      rr /= 3; cc /= 3;
      float* o = out + (((size_t)b * GHALF + rr) * GHALF + cc) * 3;
      const float* fp = feat + (size_t)b * 3 * N + n;
      atomicAdd(o + 0, fp[0]);
      atomicAdd(o + 1, fp[(size_t)N]);
      atomicAdd(o + 2, fp[(size_t)2 * N]);
    }
  }
}

extern "C" void kernel_launch(void* const* d_in, const int* in_sizes, int n_in,
                              void* d_out, int out_size, void* d_ws, size_t ws_size,
                              hipStream_t stream) {
  const float* pc1   = (const float*)d_in[0];
  const float* feat  = (const float*)d_in[1];
  const float* trans = (const float*)d_in[2];
  const int B = in_sizes[2] / 9;
  const int N = in_sizes[0] / (3 * B);   // 8192, multiple of 256
  float* out = (float*)d_out;

  int*  minbuf = (int*)d_ws;
  int2* keys   = (int2*)((char*)d_ws + (((size_t)2 * B * sizeof(int) + 255) & ~(size_t)255));

  const int nmin = 2 * B;
  const int initN = out_size > nmin ? out_size : nmin;
  init_kernel<<<(initN + 255) / 256, 256, 0, stream>>>(out, out_size, minbuf, nmin);

  const int blocksPerBatch = N / 256;
  pass1_kernel<<<B * blocksPerBatch, 256, 0, stream>>>(pc1, trans, keys, minbuf, N, blocksPerBatch);

  const long long total = (long long)B * N;
  pass2_kernel<<<(int)((total + 255) / 256), 256, 0, stream>>>(feat, keys, minbuf, out, N, B);
}